// DeepseekV3MoE_17325898072269
// MI455X (gfx1250) — compile-verified
//
#include <hip/hip_runtime.h>
#include <hip/hip_bf16.h>

// ---------------------------------------------------------------------------
// DeepseekV3 MoE forward for MI455X (gfx1250):
//   fp32 WMMA (V_WMMA_F32_16X16X4_F32) + Tensor Data Mover staging to LDS.
// ---------------------------------------------------------------------------

typedef __attribute__((ext_vector_type(2))) float        v2f;
typedef __attribute__((ext_vector_type(8))) float        v8f;
typedef __attribute__((ext_vector_type(4))) unsigned int v4u;
typedef __attribute__((ext_vector_type(4))) int          v4i;
typedef __attribute__((ext_vector_type(8))) int          v8i;

#define HD 768   // hidden size
#define ID 384   // routed intermediate
#define ISD 768  // shared intermediate
#define ED 16    // experts
#define KC 192   // K staging chunk (columns per TDM tile)
#define XP 195   // LDS pitch of a staged row: 192 + 3 TDM pad DWORDs (pitch%64==3)

#if __has_include(<hip/amd_detail/amd_gfx1250_TDM.h>)
#define TDM_SIX_ARGS 1
#endif

__device__ __forceinline__ v8f wmma_f32(v2f a, v2f b, v8f c) {
    // D = A(16x4) * B(4x16) + C(16x16), fp32
    return __builtin_amdgcn_wmma_f32_16x16x4_f32(
        false, a, false, b, (short)0, c, false, false);
}

// TDM: DMA a 16x192 fp32 tile (row stride 768 floats) from global into LDS.
// Pad: 1 DWORD every 64 DWORDs -> LDS row pitch 195 DWORDs (bank-conflict-free
// column reads for WMMA A-fragments). Element (r,c) at DWORD r*195 + c + (c>>6).
__device__ __forceinline__ void tdm_load_16x192(const float* src, unsigned lds_off,
                                                unsigned rows) {
    const unsigned long long ga = (unsigned long long)(size_t)src;
    v4u g0;
    g0[0] = 1u;                                   // count=1, user descriptor
    g0[1] = lds_off;                              // lds_addr
    g0[2] = (unsigned)(ga & 0xffffffffu);         // global_addr[31:0]
    g0[3] = (unsigned)((ga >> 32) & 0x01ffffffu)  // global_addr[56:32]
          | 0x80000000u;                          // type=2 ("image")
    v8i g1;
    g1[0] = (int)((2u << 16)        // data_size = 4 bytes
                | (1u << 20)        // pad_enable
                | (5u << 22));      // pad_interval: every 64 DWORDs (pad_amount=0 -> 1 DWORD)
    g1[1] = (int)(((unsigned)HD & 0xffffu) << 16);          // tensor_dim0 = 768 (lo16)
    g1[2] = (int)((rows & 0xffffu) << 16);                  // dim0 hi16=0 | tensor_dim1 lo16
    g1[3] = (int)(((rows >> 16) & 0xffffu) | (192u << 16)); // tensor_dim1 hi16 | tile_dim0=192
    g1[4] = 16;                                             // tile_dim1 = 16 (tile_dim2=0)
    g1[5] = HD;                                             // tensor_dim0_stride = 768
    g1[6] = 0;
    g1[7] = 0;
    const v4i z4 = {};
#ifdef TDM_SIX_ARGS
    const v8i z8 = {};
    __builtin_amdgcn_tensor_load_to_lds(g0, g1, z4, z4, z8, 0);
#else
    __builtin_amdgcn_tensor_load_to_lds(g0, g1, z4, z4, 0);
#endif
}

// ---------------------------------------------------------------------------
__global__ __launch_bounds__(256) void zero_kernel(float* __restrict__ p, int n) {
    int i = blockIdx.x * 256 + threadIdx.x;
    if (i < n) p[i] = 0.0f;
}

// ---------------------------------------------------------------------------
// Router: one wave (32 lanes) per token. Produces dense combine weights c[T,16].
__global__ __launch_bounds__(128) void router_kernel(
    const float* __restrict__ x, const float* __restrict__ rw,
    const float* __restrict__ rb, const float* __restrict__ cb,
    float* __restrict__ cOut, int T)
{
    const int token = blockIdx.x * 4 + (threadIdx.x >> 5);
    const int lane  = threadIdx.x & 31;
    if (token >= T) return;

    float acc[ED];
#pragma unroll
    for (int e = 0; e < ED; ++e) acc[e] = 0.0f;

    for (int h = lane; h < HD; h += 32) {
        const float xv = x[(size_t)token * HD + h];
#pragma unroll
        for (int e = 0; e < ED; ++e)
            acc[e] += xv * rw[(size_t)e * HD + h];
    }
#pragma unroll
    for (int off = 16; off >= 1; off >>= 1) {
#pragma unroll
        for (int e = 0; e < ED; ++e)
            acc[e] += __shfl_xor(acc[e], off, 32);
    }

    if (lane == 0) {
        float sc[ED], sf[ED];
#pragma unroll
        for (int e = 0; e < ED; ++e) {
            const float lg = acc[e] + rb[e];
            const float s  = 1.0f / (1.0f + __expf(-lg));
            sc[e] = s;
            sf[e] = s + cb[e];
        }
        float gs[4];
#pragma unroll
        for (int g = 0; g < 4; ++g) {
            float m1 = -1e30f, m2 = -1e30f;
#pragma unroll
            for (int j = 0; j < 4; ++j) {
                const float v = sf[g * 4 + j];
                if (v > m1) { m2 = m1; m1 = v; }
                else if (v > m2) { m2 = v; }
            }
            gs[g] = m1 + m2;
        }
        int g0 = 0;
#pragma unroll
        for (int g = 1; g < 4; ++g) if (gs[g] > gs[g0]) g0 = g;
        int g1 = -1;
#pragma unroll
        for (int g = 0; g < 4; ++g) {
            if (g == g0) continue;
            if (g1 < 0 || gs[g] > gs[g1]) g1 = g;
        }
        float ms[ED];
#pragma unroll
        for (int e = 0; e < ED; ++e) {
            const int g = e >> 2;
            ms[e] = (g == g0 || g == g1) ? sf[e] : 0.0f;
        }
        int   ik[4];
        float wk[4];
        float wsum = 0.0f;
#pragma unroll
        for (int k = 0; k < 4; ++k) {
            int bi = 0; float bv = -1e30f;
#pragma unroll
            for (int e = 0; e < ED; ++e)
                if (ms[e] > bv) { bv = ms[e]; bi = e; }
            ik[k] = bi; wk[k] = sc[bi];
            ms[bi] = -1e30f;
            wsum += wk[k];
        }
        const float scale = 2.5f / (wsum + 1e-20f);
        float co[ED];
#pragma unroll
        for (int e = 0; e < ED; ++e) co[e] = 0.0f;
#pragma unroll
        for (int k = 0; k < 4; ++k) co[ik[k]] = wk[k] * scale;
#pragma unroll
        for (int e = 0; e < ED; ++e)
            cOut[(size_t)token * ED + e] = co[e];
    }
}

// ---------------------------------------------------------------------------
// Routed experts, fused: grid (T/16, E). 8 waves/block.
// gate/up: 24 N-tiles (3/wave), down: 48 N-tiles (6/wave).
__global__ __launch_bounds__(256) void routed_moe_kernel(
    const float* __restrict__ x,  const float* __restrict__ c,
    const float* __restrict__ gWg, const float* __restrict__ gbg,
    const float* __restrict__ gWu, const float* __restrict__ gbu,
    const float* __restrict__ gWd, const float* __restrict__ gbd,
    float* __restrict__ out, int T)
{
    __shared__ float Xs[16 * XP];        // TDM-staged X chunk (padded pitch 195)
    __shared__ float Acts[16][ID + 1];   // act tile, pitch 385 (odd -> conflict-free)
    __shared__ float cTile[16];

    const int e    = blockIdx.y;
    const int tb   = blockIdx.x * 16;
    const int tid  = threadIdx.x;
    const int lane = tid & 31;
    const int wave = tid >> 5;
    const int m    = lane & 15;   // M row of A / N col of B / N col of C
    const int kh   = lane >> 4;   // K half: fragment K = v + 2*kh

    if (tid < 16) cTile[tid] = c[(size_t)(tb + tid) * ED + e];
    __syncthreads();
    float csum = 0.0f;
#pragma unroll
    for (int i = 0; i < 16; ++i) csum += __builtin_fabsf(cTile[i]);
    if (csum == 0.0f) return;   // uniform per block: EXEC stays all-1s for WMMA

    const float* Wg = gWg + (size_t)e * HD * ID;
    const float* Wu = gWu + (size_t)e * HD * ID;
    const unsigned xs_off = (unsigned)(size_t)(void*)Xs;

    v8f accg[3] = {};
    v8f accu[3] = {};
    const int mbase = m * XP;

    for (int ch = 0; ch < HD / KC; ++ch) {
        if (tid < 32) {   // wave 0 issues the tile DMA and drains TENSORcnt
            tdm_load_16x192(x + (size_t)tb * HD + ch * KC, xs_off, (unsigned)T);
            __builtin_amdgcn_s_wait_tensorcnt(0);
        }
        __syncthreads();
        const int kbase = ch * KC;
        for (int k0 = 0; k0 < KC; k0 += 4) {
            const int cidx = k0 + 2 * kh;
            v2f a;
            a.x = Xs[mbase + cidx + (cidx >> 6)];
            a.y = Xs[mbase + cidx + (cidx >> 6) + 1];
            const int kg = kbase + cidx;
#pragma unroll
            for (int t = 0; t < 3; ++t) {
                const int n = (wave * 3 + t) * 16 + m;
                v2f bg, bu;
                bg.x = Wg[(size_t)kg * ID + n];
                bg.y = Wg[(size_t)(kg + 1) * ID + n];
                accg[t] = wmma_f32(a, bg, accg[t]);
                bu.x = Wu[(size_t)kg * ID + n];
                bu.y = Wu[(size_t)(kg + 1) * ID + n];
                accu[t] = wmma_f32(a, bu, accu[t]);
            }
        }
        __syncthreads();   // LDS reads done before next chunk's DMA overwrites
    }

    // epilogue of gate/up: act = silu(g + bg) * (u + bu) -> LDS
#pragma unroll
    for (int t = 0; t < 3; ++t) {
        const int n  = (wave * 3 + t) * 16 + m;
        const float bg = gbg[(size_t)e * ID + n];
        const float bu = gbu[(size_t)e * ID + n];
#pragma unroll
        for (int v = 0; v < 8; ++v) {
            const int row = v + 8 * kh;
            const float g = accg[t][v] + bg;
            const float u = accu[t][v] + bu;
            Acts[row][n] = (g / (1.0f + __expf(-g))) * u;
        }
    }
    __syncthreads();

    // down GEMM: K = ID = 384, 48 output tiles over H
    v8f accd[6] = {};
    const float* Wd = gWd + (size_t)e * ID * HD;
    for (int k0 = 0; k0 < ID; k0 += 4) {
        const int kg = k0 + 2 * kh;
        v2f a;
        a.x = Acts[m][kg];
        a.y = Acts[m][kg + 1];
#pragma unroll
        for (int t = 0; t < 6; ++t) {
            const int n = (wave * 6 + t) * 16 + m;
            v2f b;
            b.x = Wd[(size_t)kg * HD + n];
            b.y = Wd[(size_t)(kg + 1) * HD + n];
            accd[t] = wmma_f32(a, b, accd[t]);
        }
    }
#pragma unroll
    for (int t = 0; t < 6; ++t) {
        const int n  = (wave * 6 + t) * 16 + m;
        const float bd = gbd[(size_t)e * HD + n];
#pragma unroll
        for (int v = 0; v < 8; ++v) {
            const int row = v + 8 * kh;
            const float cv = cTile[row];
            if (cv != 0.0f)
                atomicAdd(&out[(size_t)(tb + row) * HD + n], cv * (accd[t][v] + bd));
        }
    }
}

// ---------------------------------------------------------------------------
// Shared expert, stage 1: hs = silu(x@sWg + sbg) * (x@sWu + sbu)   [T, ISD]
__global__ __launch_bounds__(256) void shared_gateup_kernel(
    const float* __restrict__ x,
    const float* __restrict__ sWg, const float* __restrict__ sbg,
    const float* __restrict__ sWu, const float* __restrict__ sbu,
    float* __restrict__ hs, int T)
{
    __shared__ float Xs[16 * XP];

    const int tb   = blockIdx.x * 16;
    const int tid  = threadIdx.x;
    const int lane = tid & 31;
    const int wave = tid >> 5;
    const int m    = lane & 15;
    const int kh   = lane >> 4;
    const unsigned xs_off = (unsigned)(size_t)(void*)Xs;

    v8f accg[6] = {};
    v8f accu[6] = {};
    const int mbase = m * XP;

    for (int ch = 0; ch < HD / KC; ++ch) {
        if (tid < 32) {
            tdm_load_16x192(x + (size_t)tb * HD + ch * KC, xs_off, (unsigned)T);
            __builtin_amdgcn_s_wait_tensorcnt(0);
        }
        __syncthreads();
        const int kbase = ch * KC;
        for (int k0 = 0; k0 < KC; k0 += 4) {
            const int cidx = k0 + 2 * kh;
            v2f a;
            a.x = Xs[mbase + cidx + (cidx >> 6)];
            a.y = Xs[mbase + cidx + (cidx >> 6) + 1];
            const int kg = kbase + cidx;
#pragma unroll
            for (int t = 0; t < 6; ++t) {
                const int n = (wave * 6 + t) * 16 + m;
                v2f bg, bu;
                bg.x = sWg[(size_t)kg * ISD + n];
                bg.y = sWg[(size_t)(kg + 1) * ISD + n];
                accg[t] = wmma_f32(a, bg, accg[t]);
                bu.x = sWu[(size_t)kg * ISD + n];
                bu.y = sWu[(size_t)(kg + 1) * ISD + n];
                accu[t] = wmma_f32(a, bu, accu[t]);
            }
        }
        __syncthreads();
    }
#pragma unroll
    for (int t = 0; t < 6; ++t) {
        const int n  = (wave * 6 + t) * 16 + m;
        const float bg = sbg[n];
        const float bu = sbu[n];
#pragma unroll
        for (int v = 0; v < 8; ++v) {
            const int row = v + 8 * kh;
            const float g = accg[t][v] + bg;
            const float u = accu[t][v] + bu;
            hs[(size_t)(tb + row) * ISD + n] = (g / (1.0f + __expf(-g))) * u;
        }
    }
}

// ---------------------------------------------------------------------------
// Shared expert, stage 2: out += hs @ sWd + sbd
__global__ __launch_bounds__(256) void shared_down_kernel(
    const float* __restrict__ hs,
    const float* __restrict__ sWd, const float* __restrict__ sbd,
    float* __restrict__ out, int T)
{
    __shared__ float As[16 * XP];

    const int tb   = blockIdx.x * 16;
    const int tid  = threadIdx.x;
    const int lane = tid & 31;
    const int wave = tid >> 5;
    const int m    = lane & 15;
    const int kh   = lane >> 4;
    const unsigned as_off = (unsigned)(size_t)(void*)As;

    v8f acc[6] = {};
    const int mbase = m * XP;

    for (int ch = 0; ch < ISD / KC; ++ch) {
        if (tid < 32) {
            tdm_load_16x192(hs + (size_t)tb * ISD + ch * KC, as_off, (unsigned)T);
            __builtin_amdgcn_s_wait_tensorcnt(0);
        }
        __syncthreads();
        const int kbase = ch * KC;
        for (int k0 = 0; k0 < KC; k0 += 4) {
            const int cidx = k0 + 2 * kh;
            v2f a;
            a.x = As[mbase + cidx + (cidx >> 6)];
            a.y = As[mbase + cidx + (cidx >> 6) + 1];
            const int kg = kbase + cidx;
#pragma unroll
            for (int t = 0; t < 6; ++t) {
                const int n = (wave * 6 + t) * 16 + m;
                v2f b;
                b.x = sWd[(size_t)kg * HD + n];
                b.y = sWd[(size_t)(kg + 1) * HD + n];
                acc[t] = wmma_f32(a, b, acc[t]);
            }
        }
        __syncthreads();
    }
#pragma unroll
    for (int t = 0; t < 6; ++t) {
        const int n  = (wave * 6 + t) * 16 + m;
        const float bd = sbd[n];
#pragma unroll
        for (int v = 0; v < 8; ++v) {
            const int row = v + 8 * kh;
            atomicAdd(&out[(size_t)(tb + row) * HD + n], acc[t][v] + bd);
        }
    }
}

// ---------------------------------------------------------------------------
extern "C" void kernel_launch(void* const* d_in, const int* in_sizes, int n_in,
                              void* d_out, int out_size, void* d_ws, size_t ws_size,
                              hipStream_t stream) {
    const float* x   = (const float*)d_in[0];
    const float* rw  = (const float*)d_in[1];
    const float* rb  = (const float*)d_in[2];
    const float* cb  = (const float*)d_in[3];
    const float* gWg = (const float*)d_in[4];
    const float* gbg = (const float*)d_in[5];
    const float* gWu = (const float*)d_in[6];
    const float* gbu = (const float*)d_in[7];
    const float* gWd = (const float*)d_in[8];
    const float* gbd = (const float*)d_in[9];
    const float* sWg = (const float*)d_in[10];
    const float* sbg = (const float*)d_in[11];
    const float* sWu = (const float*)d_in[12];
    const float* sbu = (const float*)d_in[13];
    const float* sWd = (const float*)d_in[14];
    const float* sbd = (const float*)d_in[15];
    float* out = (float*)d_out;

    const int T = in_sizes[0] / HD;       // 2048 tokens

    float* ws = (float*)d_ws;
    float* c  = ws;                        // [T, 16]
    float* hs = ws + (size_t)T * ED;       // [T, 768]

    zero_kernel<<<(T * HD + 255) / 256, 256, 0, stream>>>(out, T * HD);
    router_kernel<<<T / 4, 128, 0, stream>>>(x, rw, rb, cb, c, T);
    routed_moe_kernel<<<dim3(T / 16, ED), 256, 0, stream>>>(
        x, c, gWg, gbg, gWu, gbu, gWd, gbd, out, T);
    shared_gateup_kernel<<<T / 16, 256, 0, stream>>>(x, sWg, sbg, sWu, sbu, hs, T);
    shared_down_kernel<<<T / 16, 256, 0, stream>>>(hs, sWd, sbd, out, T);
}